// KNN_32023276159481
// MI455X (gfx1250) — compile-verified
//
#include <hip/hip_runtime.h>

typedef __attribute__((ext_vector_type(2))) float v2f;
typedef __attribute__((ext_vector_type(8))) float v8f;

#define NPTS 4096
#define DIM  128
#define KNN  20
#define QT   16      // query rows per workgroup
#define CTW  16      // candidate cols per wave-tile
#define NW   4       // waves per workgroup
#define TSTR 132     // padded LDS tile row stride (floats)
#define TILEF (CTW * TSTR)        // floats per staged candidate tile (2112)
#define NCAND (NW * 2 * KNN)      // 160 merge candidates per row

__device__ __forceinline__ bool dless(float d1, int i1, float d2, int i2) {
    return (d1 < d2) || (d1 == d2 && i1 < i2);
}

// Fully-unrolled, constant-index sorted insert (keeps list in VGPRs).
__device__ __forceinline__ void topk_insert(float (&ld)[KNN], int (&li)[KNN],
                                            float d, int idx) {
    if (!dless(d, idx, ld[KNN - 1], li[KNN - 1])) return;
#pragma unroll
    for (int j = KNN - 1; j >= 1; --j) {
        bool keep = !dless(d, idx, ld[j], li[j]);         // d >= a[j]: slot unchanged
        bool shift = dless(d, idx, ld[j - 1], li[j - 1]); // d < a[j-1]: shift up
        float nd = keep ? ld[j] : (shift ? ld[j - 1] : d);
        int   ni = keep ? li[j] : (shift ? li[j - 1] : idx);
        ld[j] = nd; li[j] = ni;
    }
    if (dless(d, idx, ld[0], li[0])) { ld[0] = d; li[0] = idx; }
}

// Generic (flat) LDS pointer -> 32-bit LDS byte offset (shared aperture keeps
// the offset in addr[31:0]).
__device__ __forceinline__ unsigned ldsOffset(const volatile void* p) {
    return (unsigned)(uintptr_t)p;
}

// Stage one 16x128-f32 candidate tile into LDS with async B128 copies:
// 16 rows, each row = one global_load_async_to_lds_b128 (32 lanes x 16B = 512B,
// perfectly coalesced). Tracked by ASYNCcnt.
__device__ __forceinline__ void async_stage_tile(const float* __restrict__ src,
                                                 unsigned ldsBase, int lane) {
    const char* g = (const char*)src + lane * 16;       // 16B per lane
    unsigned l = ldsBase + (unsigned)lane * 16u;
#pragma unroll
    for (int r = 0; r < CTW; ++r) {
        asm volatile("global_load_async_to_lds_b128 %0, %1, off"
                     :: "v"(l), "v"(g) : "memory");
        g += DIM * 4;    // next point row (512B in memory)
        l += TSTR * 4;   // next padded LDS row (528B)
    }
}

// Kernel 1: per-point squared norms -> workspace [B*N]
__global__ __launch_bounds__(256) void knn_norms(const float* __restrict__ pc,
                                                 float* __restrict__ sq) {
    int i = blockIdx.x * blockDim.x + threadIdx.x;   // 0 .. B*N-1
    const float4* row = (const float4*)(pc + (size_t)i * DIM);
    float s = 0.f;
#pragma unroll
    for (int j = 0; j < DIM / 4; ++j) {
        float4 v = row[j];
        s += v.x * v.x + v.y * v.y + v.z * v.z + v.w * v.w;
    }
    sq[i] = s;
}

// Kernel 2: fused distance GEMM (fp32 WMMA, async-LDS double-buffered B tiles)
//           + streaming top-20 + gather.
__global__ __launch_bounds__(128) void knn_main(const float* __restrict__ pc,
                                                const float* __restrict__ sqn,
                                                float* __restrict__ out) {
    __shared__ float qTile[QT * TSTR];          // 8.4 KB  staged query tile
    __shared__ float bTile[NW * 2 * TILEF];     // 67.6 KB per-wave double buffers
    __shared__ float sqAll[NPTS];               // 16 KB   per-batch norms
    __shared__ float distT[NW * 16 * 16];       // 4 KB    per-wave C bounce tile
    __shared__ float candD[QT * NCAND];         // 10.25 KB
    __shared__ int   candI[QT * NCAND];         // 10.25 KB
    __shared__ int   selIdx[QT * KNN];          // 1.25 KB

    const int tid  = threadIdx.x;
    const int wv   = tid >> 5;
    const int lane = tid & 31;
    const int b    = blockIdx.y;
    const int n0   = blockIdx.x * QT;
    const size_t bBase = (size_t)b * NPTS;
    const float* pcb = pc + bBase * DIM;

    const int ct0 = wv * 64, ctEnd = ct0 + 64;
    const unsigned bBuf = ldsOffset(&bTile[wv * 2 * TILEF]);

    // Kick off the first candidate-tile prefetch; its latency hides behind the
    // query/norm staging below.
    async_stage_tile(pcb + (size_t)ct0 * CTW * DIM, bBuf, lane);

    // ---- stage query tile (row-padded) and batch norms into LDS ----
    for (int i = tid; i < QT * (DIM / 4); i += 128) {
        int r = i >> 5, c4 = i & 31;
        float4 v = ((const float4*)(pcb + (size_t)(n0 + r) * DIM))[c4];
        ((float4*)(qTile + r * TSTR))[c4] = v;
    }
    for (int i = tid; i < NPTS / 4; i += 128)
        ((float4*)sqAll)[i] = ((const float4*)(sqn + bBase))[i];
    __syncthreads();

    const int halfSel = (lane < 16) ? 0 : 1;  // K sub-phase per WMMA f32 layout
    const int mrow    = lane & 15;            // A row / B col index for this lane

    float sqQ[8];
#pragma unroll
    for (int v = 0; v < 8; ++v) sqQ[v] = sqAll[n0 + v + 8 * halfSel];

    // per-lane running top-20 (2 lanes per query row within this wave)
    float ld[KNN]; int li[KNN];
#pragma unroll
    for (int j = 0; j < KNN; ++j) { ld[j] = 3.4e38f; li[j] = 0x7FFFFFFF; }
    const int rloc = lane >> 1;
    const int cofs = (lane & 1) * 8;

    // ---- stream candidate tiles with double-buffered async staging ----
    for (int ct = ct0; ct < ctEnd; ++ct) {
        const int c0 = ct * CTW;
        if (ct + 1 < ctEnd) {
            // prefetch next tile into the other buffer, then retire the
            // older 16 async copies (in-order completion) for this buffer
            async_stage_tile(pcb + (size_t)(ct + 1) * CTW * DIM,
                             bBuf + (unsigned)(((ct + 1) & 1) * (TILEF * 4)), lane);
            asm volatile("s_wait_asynccnt 16" ::: "memory");
        } else {
            asm volatile("s_wait_asynccnt 0" ::: "memory");
        }
        __builtin_amdgcn_wave_barrier();

        v8f acc = {};
        const float* aptr = qTile + mrow * TSTR + 2 * halfSel;
        const float* bptr = bTile + wv * 2 * TILEF + (ct & 1) * TILEF
                            + mrow * TSTR + 2 * halfSel;
#pragma unroll 8
        for (int kk = 0; kk < DIM; kk += 4) {
            v2f a  = *(const v2f*)(aptr + kk);
            v2f bb = *(const v2f*)(bptr + kk);
            acc = __builtin_amdgcn_wmma_f32_16x16x4_f32(
                false, a, false, bb, (short)0, acc, false, false);
        }
        const float sqc = sqAll[c0 + mrow];
        // distances -> per-wave LDS bounce tile
#pragma unroll
        for (int v = 0; v < 8; ++v) {
            float dist = sqQ[v] + sqc - 2.0f * acc[v];
            distT[(wv << 8) + ((v + 8 * halfSel) << 4) + mrow] = dist;
        }
        asm volatile("s_wait_dscnt 0" ::: "memory");
        __builtin_amdgcn_wave_barrier();
        // re-map: this lane pulls 8 contiguous cols of its assigned row
        float4 d0 = *(const float4*)(&distT[(wv << 8) + (rloc << 4) + cofs]);
        float4 d1 = *(const float4*)(&distT[(wv << 8) + (rloc << 4) + cofs + 4]);
        float dv[8] = {d0.x, d0.y, d0.z, d0.w, d1.x, d1.y, d1.z, d1.w};
#pragma unroll
        for (int j = 0; j < 8; ++j)
            topk_insert(ld, li, dv[j], c0 + cofs + j);
        __builtin_amdgcn_wave_barrier();
    }

    // ---- dump per-lane lists for the cross-wave merge ----
    {
        int base = rloc * NCAND + (wv * 2 + (lane & 1)) * KNN;
#pragma unroll
        for (int j = 0; j < KNN; ++j) { candD[base + j] = ld[j]; candI[base + j] = li[j]; }
    }
    __syncthreads();

    // ---- selection: wave wv owns rows [4*wv, 4*wv+4); 20 argmin passes ----
    for (int r = wv * 4; r < wv * 4 + 4; ++r) {
        float* cd = candD + r * NCAND;
        int*   ci = candI + r * NCAND;
        for (int pass = 0; pass < KNN; ++pass) {
            float best = 3.4e38f; int bmeta = 0x7FFFFFFF;
            for (int e = lane; e < NCAND; e += 32) {      // exactly 5 iters
                float d = cd[e];
                int   m = (ci[e] << 8) | e;               // (idx, pos): stable tie-break
                if (d < best || (d == best && m < bmeta)) { best = d; bmeta = m; }
            }
#pragma unroll
            for (int off = 16; off; off >>= 1) {
                float od = __shfl_xor(best, off, 32);
                int   om = __shfl_xor(bmeta, off, 32);
                if (od < best || (od == best && om < bmeta)) { best = od; bmeta = om; }
            }
            if (lane == 0) {
                selIdx[r * KNN + pass] = bmeta >> 8;
                cd[bmeta & 255] = 3.4e38f;                // invalidate winner
            }
            asm volatile("s_wait_dscnt 0" ::: "memory");
            __builtin_amdgcn_wave_barrier();
        }
    }
    __syncthreads();

    // ---- gather neighbors: one (row,j) pair per wave iter, B128 coalesced ----
    for (int p = wv; p < QT * KNN; p += NW) {
        int r = p / KNN, j = p % KNN;
        int src = selIdx[r * KNN + j];
        const float4* s = (const float4*)(pcb + (size_t)src * DIM);
        float4* o = (float4*)(out +
            ((((size_t)b * NPTS + (n0 + r)) * KNN + j) * DIM));
        o[lane] = s[lane];                                // 32 lanes * float4 = 128 floats
    }
}

extern "C" void kernel_launch(void* const* d_in, const int* in_sizes, int n_in,
                              void* d_out, int out_size, void* d_ws, size_t ws_size,
                              hipStream_t stream) {
    (void)in_sizes; (void)n_in; (void)out_size; (void)ws_size;
    const float* pc  = (const float*)d_in[0];
    float*       out = (float*)d_out;
    float*       sqn = (float*)d_ws;          // B*N floats = 128 KB scratch

    knn_norms<<<dim3((8 * NPTS) / 256), dim3(256), 0, stream>>>(pc, sqn);
    knn_main<<<dim3(NPTS / QT, 8), dim3(128), 0, stream>>>(pc, sqn, out);
}